// Featurer_47708496724867
// MI455X (gfx1250) — compile-verified
//
#include <hip/hip_runtime.h>
#include <math.h>

// ---------------------------------------------------------------------------
// Fully fused Featurer for MI455X (gfx1250, wave32).
// One workgroup (256 threads = 8 waves) per batch element; all intermediates
// in LDS; all GEMMs via V_WMMA_F32_16X16X4_F32.
// ---------------------------------------------------------------------------

typedef __attribute__((ext_vector_type(2))) float v2f;
typedef __attribute__((ext_vector_type(4))) float v4f;
typedef __attribute__((ext_vector_type(8))) float v8f;
typedef __attribute__((ext_vector_type(2))) int   v2i;

#define SEQ      1000
#define SP       1012          // padded h row stride (bank-friendly, 16B-aligned)
#define NTILE    63            // 63*16 = 1008 covers 1000
#define NPOOL    33
#define POOLED_S 990           // 33*30
// att scale: 1 / (30*30*sqrt(33))
#define ATT_INV  1.9341967e-4f

// ---- dynamic LDS layout (float offsets) ----
#define OFF_H    0                         // h[16][SP]
#define OFF_WQ   (OFF_H + 16*SP)           // window sums q [16][36]
#define OFF_WK   (OFF_WQ + 16*36)          // window sums k [16][36]
#define OFF_RS   (OFF_WK + 16*36)          // softmax rowsum [16]
#define C1STR    952
#define OFF_C1   (OFF_RS + 16)             // conv1 out [2][16][952]
#define OFF_W2T  (OFF_C1 + 2*16*C1STR)     // conv2 weights, paired rows [80][16][2]
#define OFF_TB   (OFF_W2T + 160*16)        // t [192][12]
#define OFF_KB   (OFF_TB + 192*12)         // attention K [192][12]
#define OFF_VB   (OFF_KB + 192*12)         // attention V [192][12]
#define OFF_WB   (OFF_VB + 192*12)         // transformer layer weights (1344)
#define OFF_W1S  (OFF_WB + 1344)           // conv1 weights (102 -> 128)
#define OFF_OT   (OFF_W1S + 128)           // conv2 A-offset table (160 ints)
#define SMEM_F   (OFF_OT + 160)
#define SMEM_BYTES (SMEM_F * 4)

static __device__ __forceinline__ v8f vzero8() {
  v8f z;
#pragma unroll
  for (int i = 0; i < 8; ++i) z[i] = 0.f;
  return z;
}

static __device__ __forceinline__ v8f wmma4(v2f a, v2f b, v8f c) {
  // D = A(16x4 f32) * B(4x16 f32) + C(16x16 f32)
  return __builtin_amdgcn_wmma_f32_16x16x4_f32(false, a, false, b,
                                               (short)0, c, false, false);
}

// sum across 16 lanes (stays within each half of the wave32)
static __device__ __forceinline__ float hsum16(float v) {
  v += __shfl_xor(v, 1, 32);
  v += __shfl_xor(v, 2, 32);
  v += __shfl_xor(v, 4, 32);
  v += __shfl_xor(v, 8, 32);
  return v;
}

__global__ __launch_bounds__(256, 1) void featurer_fused(
    const float* __restrict__ x,
    const float* __restrict__ cw,   const float* __restrict__ cbias,
    const float* __restrict__ caqw, const float* __restrict__ caqb,
    const float* __restrict__ caqg, const float* __restrict__ caqe,
    const float* __restrict__ cakw, const float* __restrict__ cakb,
    const float* __restrict__ cakg, const float* __restrict__ cake,
    const float* __restrict__ capw, const float* __restrict__ capb,
    const float* __restrict__ capg, const float* __restrict__ cape,
    const float* __restrict__ w1,   const float* __restrict__ b1,
    const float* __restrict__ bng,  const float* __restrict__ bnb,
    const float* __restrict__ bnm,  const float* __restrict__ bnv,
    const float* __restrict__ w2,   const float* __restrict__ b2,
    const float* __restrict__ ln1g, const float* __restrict__ ln1b,
    const float* __restrict__ tqw,  const float* __restrict__ tqb,
    const float* __restrict__ tkw,  const float* __restrict__ tkb,
    const float* __restrict__ tvw,  const float* __restrict__ tvb,
    const float* __restrict__ tow,  const float* __restrict__ tob,
    const float* __restrict__ ln2g, const float* __restrict__ ln2b,
    const float* __restrict__ f1w,  const float* __restrict__ f1b,
    const float* __restrict__ f2w,  const float* __restrict__ f2b,
    float* __restrict__ out)
{
  extern __shared__ float sm[];
  float* hbuf = sm + OFF_H;
  float* wqs  = sm + OFF_WQ;
  float* wks  = sm + OFF_WK;
  float* rsum = sm + OFF_RS;
  float* c1   = sm + OFF_C1;
  float* w2p  = sm + OFF_W2T;
  float* tbv  = sm + OFF_TB;
  float* katt = sm + OFF_KB;
  float* vatt = sm + OFF_VB;
  float* wb   = sm + OFF_WB;
  float* w1s  = sm + OFF_W1S;
  int*   ot   = (int*)(sm + OFF_OT);

  const int tid  = threadIdx.x;
  const int lane = tid & 31;
  const int wv   = tid >> 5;          // wave id 0..7
  const int half = lane >> 4;         // 0: K=0..1 side, 1: K=2..3 side
  const int n16  = lane & 15;         // N column / M row within half
  const int b    = blockIdx.x;

  // =========================================================================
  // Phase 1: h = (x @ change_w + change_b)^T   -> hbuf[c][s]
  // WMMA tiles: M=16 (s), N=16 (c), K=22 padded to 24 (6 k-steps).
  // A needs no masking: K-pad columns hit zeroed B rows, and out-of-range
  // s rows only produce D rows that are never consumed downstream.
  // =========================================================================
  v2f cwf[6];
#pragma unroll
  for (int kk = 0; kk < 6; ++kk) {
    int kbs = kk * 4 + half * 2;
    cwf[kk].x = (kbs     < 22) ? cw[(kbs    ) * 16 + n16] : 0.f;
    cwf[kk].y = (kbs + 1 < 22) ? cw[(kbs + 1) * 16 + n16] : 0.f;
  }
  const float cbn = cbias[n16];

  for (int tile = wv; tile < NTILE; tile += 8) {
    const int s0   = tile * 16;
    const int srow = s0 + n16;
    const int sc   = srow < SEQ ? srow : (SEQ - 1);     // address clamp only
    const float* xr = x + ((size_t)b * SEQ + sc) * 22;
    __builtin_prefetch(xr + 16 * 22, 0, 1);             // -> global_prefetch_b8
    v8f acc = vzero8();
#pragma unroll
    for (int kk = 0; kk < 6; ++kk) {
      const int kbs = kk * 4 + half * 2;
      const int kc  = (kk == 5) ? 20 : kbs;             // clamp; B rows >=22 are 0
      const v2f a   = *(const v2f*)(xr + kc);           // aligned global_load_b64
      acc = wmma4(a, cwf[kk], acc);
    }
    // rows of each half are s-contiguous -> two ds_store_b128
    float* hp = &hbuf[n16 * SP + s0 + half * 8];
    v4f lo, hi;
#pragma unroll
    for (int r = 0; r < 4; ++r) { lo[r] = acc[r] + cbn; hi[r] = acc[r + 4] + cbn; }
    *(v4f*)hp       = lo;
    *(v4f*)(hp + 4) = hi;
  }
  __syncthreads();

  // =========================================================================
  // Phase 2: 4x ChannelAttention with residual.
  //   q/k GEMMs feed only per-window sums (pool factorizes through the dot
  //   product); softmax rowsum is folded into the projection weight rows.
  // =========================================================================
  v2f qwf[4], kwf[4], pwf[4];
#pragma unroll
  for (int kk = 0; kk < 4; ++kk) {
    int kbs = kk * 4 + half * 2;
    qwf[kk].x = caqw[kbs * 16 + n16]; qwf[kk].y = caqw[(kbs + 1) * 16 + n16];
    kwf[kk].x = cakw[kbs * 16 + n16]; kwf[kk].y = cakw[(kbs + 1) * 16 + n16];
    pwf[kk].x = capw[kbs * 16 + n16]; pwf[kk].y = capw[(kbs + 1) * 16 + n16];
  }
  const float qbn = caqb[n16], qgn = caqg[n16], qen = caqe[n16];
  const float kbn = cakb[n16], kgn = cakg[n16], ken = cake[n16];
  const float pbn = capb[n16], pgn = capg[n16], pen = cape[n16];

  for (int it = 0; it < 4; ++it) {
    for (int i = tid; i < 16 * 36 * 2; i += 256) wqs[i] = 0.f;
    __syncthreads();

    // ---- q and k GEMMs + LN + window-sum accumulation ----
    for (int tile = wv; tile < NTILE; tile += 8) {
      const int s0 = tile * 16;
      v8f aq = vzero8(), ak = vzero8();
#pragma unroll
      for (int kk = 0; kk < 4; ++kk) {
        int kbs = kk * 4 + half * 2;
        v2f a;
        a.x = hbuf[(kbs    ) * SP + s0 + n16];
        a.y = hbuf[(kbs + 1) * SP + s0 + n16];
        aq = wmma4(a, qwf[kk], aq);
        ak = wmma4(a, kwf[kk], ak);
      }
#pragma unroll
      for (int r = 0; r < 8; ++r) {
        const int sr = s0 + r + half * 8;
        // LN over channels (16 lanes) for q
        float vq = aq[r] + qbn;
        float mq = hsum16(vq) * 0.0625f;
        float dq = vq - mq;
        float yq = dq * rsqrtf(hsum16(dq * dq) * 0.0625f + 1e-5f) * qgn + qen;
        // LN for k
        float vk = ak[r] + kbn;
        float mk = hsum16(vk) * 0.0625f;
        float dk = vk - mk;
        float yk = dk * rsqrtf(hsum16(dk * dk) * 0.0625f + 1e-5f) * kgn + ken;
        if (sr < POOLED_S) {
          const unsigned p = (unsigned)sr / 30u;
          atomicAdd(&wqs[n16 * 36 + p], yq);
          atomicAdd(&wks[n16 * 36 + p], yk);
        }
      }
    }
    __syncthreads();

    // ---- 16x16 attention + softmax + rowsum (threads 0..15) ----
    if (tid < 16) {
      float e[16];
      float mx = -1e30f;
      for (int m = 0; m < 16; ++m) {
        float s = 0.f;
        for (int p = 0; p < NPOOL; ++p) s += wqs[tid * 36 + p] * wks[m * 36 + p];
        e[m] = s * ATT_INV;
        mx = fmaxf(mx, e[m]);
      }
      float sum = 0.f;
      for (int m = 0; m < 16; ++m) { e[m] = __expf(e[m] - mx); sum += e[m]; }
      const float isum = 1.f / sum;
      float rs = 0.f;
      for (int m = 0; m < 16; ++m) rs += e[m] * isum;   // == sum_m softmax
      rsum[tid] = rs;
    }
    __syncthreads();

    // ---- projection GEMM (B rows scaled by rowsum) + LN + residual ----
    for (int tile = wv; tile < NTILE; tile += 8) {
      const int s0 = tile * 16;
      v8f acc = vzero8();
#pragma unroll
      for (int kk = 0; kk < 4; ++kk) {
        int kbs = kk * 4 + half * 2;
        v2f a, bb;
        a.x  = hbuf[(kbs    ) * SP + s0 + n16];
        a.y  = hbuf[(kbs + 1) * SP + s0 + n16];
        bb.x = pwf[kk].x * rsum[kbs];
        bb.y = pwf[kk].y * rsum[kbs + 1];
        acc = wmma4(a, bb, acc);
      }
      float yv[8];
#pragma unroll
      for (int r = 0; r < 8; ++r) {
        float vp = acc[r] + pbn;
        float mp = hsum16(vp) * 0.0625f;
        float dp = vp - mp;
        yv[r] = dp * rsqrtf(hsum16(dp * dp) * 0.0625f + 1e-5f) * pgn + pen;
      }
      // vector read-modify-write of the residual (b128 LDS ops)
      float* hp = &hbuf[n16 * SP + s0 + half * 8];
      v4f o0 = *(v4f*)hp;
      v4f o1 = *(v4f*)(hp + 4);
#pragma unroll
      for (int r = 0; r < 4; ++r) { o0[r] += yv[r]; o1[r] += yv[r + 4]; }
      *(v4f*)hp       = o0;
      *(v4f*)(hp + 4) = o1;
    }
    __syncthreads();
  }

  // =========================================================================
  // Phase 3: conv1 (51-tap) + BN + LeakyReLU(0.2) -> c1[oc][c][s]
  // Also stage conv2 weights (paired-row layout) and A-offset table.
  // =========================================================================
  for (int i = tid; i < 102; i += 256) w1s[i] = w1[i];
  for (int i = tid; i < 160 * 16; i += 256) {
    // w2p[pair][n][0:1] = B[2*pair + 0:1][n],  B[k][e] = w2[e*160 + k]
    const int p = i >> 5, r = i & 31, n = r >> 1, kbit = i & 1;
    const int k = p * 2 + kbit;
    w2p[i] = (n < 10) ? w2[n * 160 + k] : 0.f;
  }
  for (int i = tid; i < 160; i += 256) {
    const int oc = i / 80, rem = i - oc * 80, c = rem / 5, t = rem - c * 5;
    ot[i] = (oc * 16 + c) * C1STR + t;
  }
  __syncthreads();

  for (int idx = tid; idx < 2 * 16 * 950; idx += 256) {
    const int oc  = idx / 15200;
    const int rem = idx - oc * 15200;
    const int c   = rem / 950;
    const int s   = rem - c * 950;
    float a = b1[oc];
    const float* hr = &hbuf[c * SP + s];
    const float* wr = &w1s[oc * 51];
#pragma unroll 17
    for (int t = 0; t < 51; ++t) a += hr[t] * wr[t];
    a = (a - bnm[oc]) * rsqrtf(bnv[oc] + 1e-5f) * bng[oc] + bnb[oc];
    a = a > 0.f ? a : 0.2f * a;
    c1[(oc * 16 + c) * C1STR + s] = a;
  }
  __syncthreads();

  // =========================================================================
  // Phase 4: conv2 (stride 5 == kernel 5 -> exact GEMM [190,160]@[160,10])
  // WMMA: M=16 (j), N=16 (e, 10 valid), K=160 (40 k-steps, fully unrolled).
  // A offsets come from the LDS table (ds_load_b64 per step, immediate offs).
  // =========================================================================
  const float b2n = (n16 < 10) ? b2[n16] : 0.f;
  for (int tile = wv; tile < 12; tile += 8) {
    const int j0 = tile * 16;
    const int j  = j0 + n16;
    const int jc = j < 190 ? j : 189;                  // address clamp only
    const float* c1j = c1 + 5 * jc;
    const v2i* otp   = (const v2i*)ot + half;          // pair index = kk*2+half
    const float* wbp = w2p + half * 32 + n16 * 2;
    v8f acc = vzero8();
#pragma unroll
    for (int kk = 0; kk < 40; ++kk) {
      const v2i o  = otp[kk * 2];                      // ds_load_b64, imm offset
      const v2f bb = *(const v2f*)(wbp + kk * 64);     // ds_load_b64, imm offset
      v2f a;
      a.x = c1j[o.x];
      a.y = c1j[o.y];
      acc = wmma4(a, bb, acc);
    }
#pragma unroll
    for (int r = 0; r < 8; ++r) {
      const int jr = j0 + r + half * 8;
      if (jr < 190 && n16 < 10) tbv[jr * 12 + n16] = acc[r] + b2n;
    }
  }
  __syncthreads();

  // =========================================================================
  // Phase 5: 3-layer transformer on [190,10]; thread = sequence row.
  // =========================================================================
  float tr[10];
  const int  row = tid;
  const bool rv  = row < 190;
  if (rv) {
#pragma unroll
    for (int e = 0; e < 10; ++e) tr[e] = tbv[row * 12 + e];
  }

  for (int L = 0; L < 3; ++L) {
    // stage this layer's weights in LDS
    for (int i = tid; i < 100; i += 256) {
      wb[i]       = tqw[L * 100 + i];
      wb[100 + i] = tkw[L * 100 + i];
      wb[200 + i] = tvw[L * 100 + i];
      wb[300 + i] = tow[L * 100 + i];
    }
    for (int i = tid; i < 400; i += 256) {
      wb[400 + i] = f1w[L * 400 + i];
      wb[800 + i] = f2w[L * 400 + i];
    }
    if (tid < 10) {
      wb[1200 + tid] = tqb[L * 10 + tid];  wb[1210 + tid] = tkb[L * 10 + tid];
      wb[1220 + tid] = tvb[L * 10 + tid];  wb[1230 + tid] = tob[L * 10 + tid];
      wb[1240 + tid] = ln1g[L * 10 + tid]; wb[1250 + tid] = ln1b[L * 10 + tid];
      wb[1260 + tid] = ln2g[L * 10 + tid]; wb[1270 + tid] = ln2b[L * 10 + tid];
      wb[1320 + tid] = f2b[L * 10 + tid];
    }
    if (tid < 40) wb[1280 + tid] = f1b[L * 40 + tid];
    __syncthreads();

    float q[10];
    if (rv) {
      // LN1
      float m = 0.f;
#pragma unroll
      for (int e = 0; e < 10; ++e) m += tr[e];
      m *= 0.1f;
      float var = 0.f;
#pragma unroll
      for (int e = 0; e < 10; ++e) { float d = tr[e] - m; var += d * d; }
      const float rs_ = rsqrtf(var * 0.1f + 1e-5f);
      float y[10];
#pragma unroll
      for (int e = 0; e < 10; ++e) y[e] = (tr[e] - m) * rs_ * wb[1240 + e] + wb[1250 + e];
      // q, k, v
      float kk2[10], vv[10];
#pragma unroll
      for (int o = 0; o < 10; ++o) { q[o] = wb[1200 + o]; kk2[o] = wb[1210 + o]; vv[o] = wb[1220 + o]; }
      for (int in = 0; in < 10; ++in) {
        const float yv = y[in];
#pragma unroll
        for (int o = 0; o < 10; ++o) {
          q[o]   += yv * wb[in * 10 + o];
          kk2[o] += yv * wb[100 + in * 10 + o];
          vv[o]  += yv * wb[200 + in * 10 + o];
        }
      }
      const float qs = 0.316227766016838f;  // 1/sqrt(emb=10)
#pragma unroll
      for (int o = 0; o < 10; ++o) q[o] *= qs;
#pragma unroll
      for (int o = 0; o < 10; ++o) { katt[row * 12 + o] = kk2[o]; vatt[row * 12 + o] = vv[o]; }
    }
    __syncthreads();

    if (rv) {
      // online-softmax attention, 5 heads of dim 2
      float mx[5], sme[5], a0[5], a1[5];
#pragma unroll
      for (int h = 0; h < 5; ++h) { mx[h] = -1e30f; sme[h] = 0.f; a0[h] = 0.f; a1[h] = 0.f; }
      for (int jj = 0; jj < 190; ++jj) {
#pragma unroll
        for (int h = 0; h < 5; ++h) {
          const float e = q[2 * h]     * katt[jj * 12 + 2 * h]
                        + q[2 * h + 1] * katt[jj * 12 + 2 * h + 1];
          const float nm   = fmaxf(mx[h], e);
          const float corr = __expf(mx[h] - nm);
          const float p    = __expf(e - nm);
          sme[h] = sme[h] * corr + p;
          a0[h]  = a0[h]  * corr + p * vatt[jj * 12 + 2 * h];
          a1[h]  = a1[h]  * corr + p * vatt[jj * 12 + 2 * h + 1];
          mx[h]  = nm;
        }
      }
      float ao[10];
#pragma unroll
      for (int h = 0; h < 5; ++h) {
        const float is = 1.f / sme[h];
        ao[2 * h] = a0[h] * is; ao[2 * h + 1] = a1[h] * is;
      }
      // output projection + residual
#pragma unroll
      for (int e = 0; e < 10; ++e) {
        float s = wb[1230 + e];
        for (int in = 0; in < 10; ++in) s += ao[in] * wb[300 + in * 10 + e];
        tr[e] += s;
      }
      // LN2
      float m = 0.f;
#pragma unroll
      for (int e = 0; e < 10; ++e) m += tr[e];
      m *= 0.1f;
      float var = 0.f;
#pragma unroll
      for (int e = 0; e < 10; ++e) { float d = tr[e] - m; var += d * d; }
      const float rs2 = rsqrtf(var * 0.1f + 1e-5f);
      float y2[10];
#pragma unroll
      for (int e = 0; e < 10; ++e) y2[e] = (tr[e] - m) * rs2 * wb[1260 + e] + wb[1270 + e];
      // FF: 10 -> 40 (exact GELU) -> 10, residual
      float g1v[40];
      for (int o = 0; o < 40; ++o) {
        float s = wb[1280 + o];
#pragma unroll
        for (int in = 0; in < 10; ++in) s += y2[in] * wb[400 + in * 40 + o];
        g1v[o] = 0.5f * s * (1.f + erff(s * 0.70710678118654752f));
      }
#pragma unroll
      for (int e = 0; e < 10; ++e) tr[e] += wb[1320 + e];
      for (int o = 0; o < 40; ++o) {
        const float gv = g1v[o];
#pragma unroll
        for (int e = 0; e < 10; ++e) tr[e] += gv * wb[800 + o * 10 + e];
      }
    }
    __syncthreads();
  }

  if (rv) {
#pragma unroll
    for (int e = 0; e < 10; ++e) out[((size_t)b * 190 + row) * 10 + e] = tr[e];
  }
}

extern "C" void kernel_launch(void* const* d_in, const int* in_sizes, int n_in,
                              void* d_out, int out_size, void* d_ws, size_t ws_size,
                              hipStream_t stream) {
  (void)n_in; (void)out_size; (void)d_ws; (void)ws_size;
  const float* p[39];
  for (int i = 0; i < 39; ++i) p[i] = (const float*)d_in[i];
  const int nb = in_sizes[0] / (SEQ * 22);   // 512

  hipFuncSetAttribute((const void*)featurer_fused,
                      hipFuncAttributeMaxDynamicSharedMemorySize, SMEM_BYTES);

  featurer_fused<<<nb, 256, SMEM_BYTES, stream>>>(
      p[0],                                  // x
      p[1], p[2],                            // change_w, change_b
      p[3], p[4], p[5], p[6],                // ca_q
      p[7], p[8], p[9], p[10],               // ca_k
      p[11], p[12], p[13], p[14],            // ca_p
      p[15], p[16],                          // conv1
      p[17], p[18], p[19], p[20],            // bn
      p[21], p[22],                          // conv2
      p[23], p[24],                          // tf_ln1
      p[25], p[26], p[27], p[28],            // tf_q, tf_k
      p[29], p[30], p[31], p[32],            // tf_v, tf_o
      p[33], p[34],                          // tf_ln2
      p[35], p[36], p[37], p[38],            // tf_ff1, tf_ff2
      (float*)d_out);
}